// OBFNet_24756191494235
// MI455X (gfx1250) — compile-verified
//
#include <hip/hip_runtime.h>

typedef float v2f __attribute__((ext_vector_type(2)));
typedef float v8f __attribute__((ext_vector_type(8)));

#define MIC   4
#define FD    31
#define LH    512
#define WIN   128
#define FRM   1000
#define BATCH 8
#define LENF  (WIN + FD - 1)        // 158
#define XPAD  (LH - FD)             // 481 zeros of left pad
#define NGRP  (BATCH * MIC * FRM)   // 32000 groups
#define KPAD  32                    // FD padded to 32 (row 31 zeroed)

// ---------------------------------------------------------------------------
// Kernel 1: build AB_w [mic, KPAD, LH] in workspace (row 31 zeroed).
// One wave per mic; 512-step matrix-power recurrence with LDS double buffer.
// ---------------------------------------------------------------------------
__global__ void obf_build_abw(const float* __restrict__ Apole,
                              float* __restrict__ abw) {
  const int m = blockIdx.x;
  const int i = threadIdx.x;           // row index 0..31 (31 rows valid)
  __shared__ float ap_s[32], ss_s[32];
  __shared__ float vbuf[2][32];

  float a0 = 0.f;
  if (i < FD) {
    a0 = Apole[m * FD + i];
    a0 = fminf(fmaxf(a0, 0.0f), 0.9999f);   // hardtanh(0, 0.9999)
  }
  ap_s[i] = (i < FD) ? a0 : 0.f;
  ss_s[i] = (i < FD) ? sqrtf(1.f - a0 * a0) : 0.f;
  __syncthreads();

  // Row i of A: A[i][j] = s_i s_j * prod_{t=j+1..i-1}(-ap_t) for j<i; ap_i at j==i.
  // B[i] = s_i * prod_{t<i}(-ap_t).
  float Arow[FD];
  float Bi = 0.f;
  if (i < FD) {
    const float si = ss_s[i];
    for (int j = 0; j < FD; ++j) Arow[j] = 0.f;
    float p = 1.f;
    for (int j = i - 1; j >= 0; --j) {
      Arow[j] = si * ss_s[j] * p;
      p *= -ap_s[j];
    }
    Arow[i] = ap_s[i];
    Bi = si * p;
  }
  vbuf[0][i] = (i < FD) ? Bi : 0.f;
  __syncthreads();

  float* wsm = abw + m * KPAD * LH;
  int cur = 0;
  for (int k = 0; k < LH; ++k) {
    if (i < FD) wsm[i * LH + (LH - 1 - k)] = vbuf[cur][i];  // flip: ys[k] -> slot LH-1-k
    float nv = 0.f;
    if (i < FD) {
      for (int j = 0; j < FD; ++j) nv += Arow[j] * vbuf[cur][j];
    }
    vbuf[cur ^ 1][i] = nv;
    __syncthreads();
    cur ^= 1;
  }
  for (int l = i; l < LH; l += 32) wsm[FD * LH + l] = 0.f;   // zero K-pad row
}

// ---------------------------------------------------------------------------
// Kernel 2: h[bm, f, l] = sum_k filt[bm, f, k] * AB_w[m, k, l]
// One block = one (bm, f-tile); 8 waves cover all 512 N columns (64 each).
// ft/bm/f0 derive from blockIdx only -> store predicates are SCALAR: the
// full-tile path (62/63 blocks) has zero exec masking. Partial tile (f0=992)
// reduces its f<1000 guard to the single predicate half==0.
// A loads branch-free: row index clamped (garbage rows never stored); the
// k=31 column needs no guard because AB_w's pad row 31 is exactly zero.
// ---------------------------------------------------------------------------
__global__ __launch_bounds__(256) void obf_h_gemm(const float* __restrict__ filt,
                                                  const float* __restrict__ abw,
                                                  float* __restrict__ hout) {
  const int lane = threadIdx.x & 31;
  const int lg   = threadIdx.x >> 5;      // 64-column strip index (wave id)
  const int ft   = blockIdx.x % 63;       // scalar
  const int bm   = blockIdx.x / 63;       // scalar, b*4 + m
  const int f0   = ft * 16;               // scalar
  const int half = lane >> 4;             // lane half selects K pair
  const int r    = lane & 15;             // M row (A) / N col (B, D)

  const int fr  = f0 + r;
  const int frc = (fr < FRM) ? fr : (FRM - 1);          // clamp: no exec branches
  const float* fbase = filt + (size_t)bm * FRM * FD + (size_t)frc * FD;

  // Preload A fragments for all 8 K-steps (reused across 4 N-tiles).
  v2f Afrag[8];
#pragma unroll
  for (int kk = 0; kk < 8; ++kk) {
    const int ka = kk * 4 + 2 * half;     // <= 30, always valid
    int kb = ka + 1;                      // == 31 only at kk==7,half==1
    if (kb > FD - 1) kb = FD - 1;         // value irrelevant: B row 31 is zero
    Afrag[kk].x = fbase[ka];
    Afrag[kk].y = fbase[kb];
  }

  const float* bcol = abw + (size_t)(bm & 3) * KPAD * LH + r;
  float* obase = hout + (size_t)bm * FRM * LH + (size_t)f0 * LH + r;

#pragma unroll
  for (int lg2 = 0; lg2 < 4; ++lg2) {
    const int l0 = (lg * 4 + lg2) * 16;
    const float* bb = bcol + l0;
    v8f acc = {};
#pragma unroll
    for (int kk = 0; kk < 8; ++kk) {
      const int ka = kk * 4 + 2 * half;
      v2f B;
      B.x = bb[(size_t)ka * LH];
      B.y = bb[(size_t)(ka + 1) * LH];    // row 31 = zero pad
      acc = __builtin_amdgcn_wmma_f32_16x16x4_f32(false, Afrag[kk], false, B,
                                                  (short)0, acc, false, false);
    }
    // D: lane r = col l, VGPR vi + 8*half = row f0 + vi + 8*half
    float* o = obase + l0 + (size_t)(8 * half) * LH;
    if (ft != 62) {                       // scalar branch: full 16-row tile
#pragma unroll
      for (int vi = 0; vi < 8; ++vi) o[(size_t)vi * LH] = acc[vi];
    } else {                              // f0=992: rows valid iff half==0
      if (half == 0) {
#pragma unroll
        for (int vi = 0; vi < 8; ++vi) o[(size_t)vi * LH] = acc[vi];
      }
    }
  }
}

// ---------------------------------------------------------------------------
// Kernel 3: grouped correlation via WMMA.
// out[g, 16n+m] = sum_{e,d} x_pad[g, 16e+m+d] * h[g, 16(e-n)+d]
// 481-sample zero pad restricts e to [29, 38] (10 blocks instead of 40).
// h staged in LDS with a zeroed 128-float tail so shifted reads for
// e-n > 31 land on zeros -> no selects in the inner loop.
// ---------------------------------------------------------------------------
__global__ __launch_bounds__(256) void obf_conv(const float* __restrict__ x,
                                                const float* __restrict__ h,
                                                float* __restrict__ out) {
  __shared__ float xs[8][176];                      // x_pad[464..639) per wave
  __shared__ __align__(16) float hs[8][LH + 128];   // 512 taps + zero tail
  const int lane = threadIdx.x & 31;
  const int wave = threadIdx.x >> 5;
  const int g = blockIdx.x * 8 + wave;

  const float*  xg  = x + (size_t)g * LENF;
  const float4* hg4 = (const float4*)(h + (size_t)g * LH);  // 2048B-aligned
  for (int i = lane; i < 176; i += 32) {
    const int idx = 464 + i;
    xs[wave][i] = (idx >= XPAD && idx < XPAD + LENF) ? xg[idx - XPAD] : 0.f;
  }
  float4* hv = (float4*)&hs[wave][0];
  for (int i = lane; i < LH / 4; i += 32) hv[i] = hg4[i];
  for (int i = lane; i < 128; i += 32) hs[wave][LH + i] = 0.f;
  __syncthreads();

  const int half = lane >> 4;
  const int r = lane & 15;       // A: row m ; B/D: col n
  v8f acc = {};
#pragma unroll
  for (int e = 29; e <= 38; ++e) {
    const int xb = 16 * e - 464 + r;    // window base for row m=r
    const int hb = 16 * (e - r);        // h block base for col n=r (zero tail if >511)
#pragma unroll
    for (int q = 0; q < 4; ++q) {
      const int d0 = 4 * q + 2 * half;
      v2f A, B;
      A.x = xs[wave][xb + d0];
      A.y = xs[wave][xb + d0 + 1];
      B.x = hs[wave][hb + d0];
      B.y = hs[wave][hb + d0 + 1];
      acc = __builtin_amdgcn_wmma_f32_16x16x4_f32(false, A, false, B,
                                                  (short)0, acc, false, false);
    }
  }
  // D[m=vi+8*half, n=r]; only n<8 are real outputs (WIN=128=8x16)
  if (r < 8) {
    float* og = out + (size_t)g * WIN + 16 * r + 8 * half;  // 32B-aligned
    float4 lo = {acc[0], acc[1], acc[2], acc[3]};
    float4 hi = {acc[4], acc[5], acc[6], acc[7]};
    ((float4*)og)[0] = lo;
    ((float4*)og)[1] = hi;
  }
}

// ---------------------------------------------------------------------------
extern "C" void kernel_launch(void* const* d_in, const int* in_sizes, int n_in,
                              void* d_out, int out_size, void* d_ws, size_t ws_size,
                              hipStream_t stream) {
  (void)in_sizes; (void)n_in; (void)out_size; (void)ws_size;
  const float* input      = (const float*)d_in[0];  // [8,4,1000,158]
  const float* all_filter = (const float*)d_in[1];  // [8,4,1,1000,31]
  const float* Apole      = (const float*)d_in[2];  // [4,31]

  float* bf_out = (float*)d_out;                       // [32000,128]
  float* h_out  = bf_out + (size_t)NGRP * WIN;         // [32000,512] (tuple elem 2)
  float* abw    = (float*)d_ws;                        // [4,32,512] = 256 KB

  obf_build_abw<<<dim3(MIC), dim3(32), 0, stream>>>(Apole, abw);
  obf_h_gemm<<<dim3(63 * 32), dim3(256), 0, stream>>>(all_filter, abw, h_out);
  obf_conv<<<dim3(NGRP / 8), dim3(256), 0, stream>>>(input, h_out, bf_out);
}